// DAGNNConv_1846835938000
// MI455X (gfx1250) — compile-verified
//
#include <hip/hip_runtime.h>
#include <math.h>

typedef __attribute__((ext_vector_type(2))) float v2f;
typedef __attribute__((ext_vector_type(8))) float v8f;

#define DIM   64
#define KHOPS 10

// ---------------------------------------------------------------- zero fill
__global__ void k_zero(float* __restrict__ p, long n) {
    long i  = (long)blockIdx.x * blockDim.x + threadIdx.x;
    long st = (long)gridDim.x * blockDim.x;
    for (; i < n; i += st) p[i] = 0.0f;
}

// ---------------------------------------------------------------- in-degree
__global__ void k_deg(const int* __restrict__ dst, int E, float* __restrict__ deg) {
    long i  = (long)blockIdx.x * blockDim.x + threadIdx.x;
    long st = (long)gridDim.x * blockDim.x;
    for (; i < E; i += st) unsafeAtomicAdd(&deg[dst[i]], 1.0f);
}

// ---------------------------------------------------------------- deg -> deg^-0.5
__global__ void k_rsqrt(float* __restrict__ p, int n) {
    int i = blockIdx.x * blockDim.x + threadIdx.x;
    if (i < n) p[i] = rsqrtf(p[i]);
}

// ---------------------------------------------------------------- SpMM hop:
// RAW[dst] += h[src] * norm[src]   (one wave32 per edge, float2 per lane)
__global__ void __launch_bounds__(256)
k_scatter(const float* __restrict__ h, const float* __restrict__ nrm,
          const int* __restrict__ src, const int* __restrict__ dst,
          int E, float* __restrict__ acc) {
    int  lane = threadIdx.x & 31;
    long w    = (long)blockIdx.x * (blockDim.x >> 5) + (threadIdx.x >> 5);
    long nw   = (long)gridDim.x * (blockDim.x >> 5);
    for (long e = w; e < E; e += nw) {
        int   sN = src[e];
        int   dN = dst[e];
        float ns = nrm[sN];
        float2 v = ((const float2*)(h + (size_t)sN * DIM))[lane];
        float* ap = acc + (size_t)dN * DIM + lane * 2;
        unsafeAtomicAdd(ap,     v.x * ns);
        unsafeAtomicAdd(ap + 1, v.y * ns);
    }
}

// ---------------------------------------------------------------- fused gate:
// per 16-node tile (one wave32):
//   dotRaw = hin . s  via 16x V_WMMA_F32_16X16X4_F32 (raw values; norm folded
//   into the sigmoid argument since dot(h*nm, s) == nm * dot(h, s))
//   hpost = hin * norm ; out += sigmoid(nm*dot) * hpost ; POST = hpost ; RAW = 0
__global__ void __launch_bounds__(256)
k_gate(float* __restrict__ hin, const float* __restrict__ nrm,
       const float* __restrict__ sv, float* __restrict__ hstore,
       float* __restrict__ out, int nNodes, int applyNorm, int zeroIn) {
    int  lane = threadIdx.x & 31;
    long w    = (long)blockIdx.x * (blockDim.x >> 5) + (threadIdx.x >> 5);
    long nw   = (long)gridDim.x * (blockDim.x >> 5);
    long T    = ((long)nNodes + 15) / 16;

    int ko = (lane >> 4) * 2;                      // K sub-offset per lane half

    // B operand: s-chunk broadcast across all 16 columns; loaded once per wave.
    float2 b[16];
    #pragma unroll
    for (int i = 0; i < 16; ++i)
        b[i] = *(const float2*)(sv + i * 4 + ko);

    for (long t = w; t < T; t += nw) {             // uniform per wave: EXEC stays full
        int nb   = (int)t * 16;
        int rowA = nb + (lane & 15);               // A-operand row for this lane
        int rowC = rowA < nNodes ? rowA : (nNodes - 1);  // clamp; garbage rows unused
        const float* hr = hin + (size_t)rowC * DIM;

        // Stage all 16 A chunks first so loads issue as a clause and overlap WMMA.
        float2 a[16];
        #pragma unroll
        for (int i = 0; i < 16; ++i)
            a[i] = *(const float2*)(hr + i * 4 + ko);

        v8f d = {0.f, 0.f, 0.f, 0.f, 0.f, 0.f, 0.f, 0.f};
        #pragma unroll
        for (int i = 0; i < 16; ++i) {
            v2f A = {a[i].x, a[i].y};
            v2f B = {b[i].x, b[i].y};
            d = __builtin_amdgcn_wmma_f32_16x16x4_f32(
                    false, A, false, B, (short)0, d, false, false);
        }

        // D layout: VGPR r of lane l holds M = r + 8*(l/16)  -> gates are local.
        int half = lane >> 4;
        int c    = lane & 15;                      // 4-float dim chunk per lane
        #pragma unroll
        for (int r = 0; r < 8; ++r) {
            int m = nb + half * 8 + r;
            if (m < nNodes) {
                float nm = applyNorm ? nrm[m] : 1.0f;
                float g  = 1.0f / (1.0f + __expf(-nm * d[r]));
                float4 v = ((const float4*)(hin + (size_t)m * DIM))[c];
                v.x *= nm; v.y *= nm; v.z *= nm; v.w *= nm;
                float4* op = (float4*)(out + (size_t)m * DIM) + c;
                float4 o = *op;
                o.x += g * v.x; o.y += g * v.y; o.z += g * v.z; o.w += g * v.w;
                *op = o;
                if (hstore) ((float4*)(hstore + (size_t)m * DIM))[c] = v;
                if (zeroIn) {
                    float4 z = {0.f, 0.f, 0.f, 0.f};
                    ((float4*)(hin + (size_t)m * DIM))[c] = z;
                }
            }
        }
    }
}

// ---------------------------------------------------------------- launcher
static inline int nblk(long threads, int tb) {
    long b = (threads + tb - 1) / tb;
    if (b < 1) b = 1;
    if (b > 1048576) b = 1048576;
    return (int)b;
}

extern "C" void kernel_launch(void* const* d_in, const int* in_sizes, int n_in,
                              void* d_out, int out_size, void* d_ws, size_t ws_size,
                              hipStream_t stream) {
    const float* feats = (const float*)d_in[0];
    const float* s     = (const float*)d_in[1];
    const int*   src   = (const int*)d_in[2];
    const int*   dst   = (const int*)d_in[3];
    float*       out   = (float*)d_out;

    int N = in_sizes[0] / DIM;
    int E = in_sizes[2];

    float* ws   = (float*)d_ws;
    float* nrm  = ws;                         // [N]      degrees -> deg^-0.5
    float* RAW  = ws + N;                     // [N*DIM]  scatter accumulation target
    float* POST = RAW + (size_t)N * DIM;      // [N*DIM]  post-scaled h_k

    const int tb = 256;

    k_zero<<<nblk((long)N, tb), tb, 0, stream>>>(nrm, N);
    k_zero<<<nblk((long)N * DIM, tb), tb, 0, stream>>>(RAW, (long)N * DIM);
    k_zero<<<nblk((long)out_size, tb), tb, 0, stream>>>(out, (long)out_size);

    k_deg<<<nblk((long)E, tb), tb, 0, stream>>>(dst, E, nrm);
    k_rsqrt<<<nblk((long)N, tb), tb, 0, stream>>>(nrm, N);

    long T = ((long)N + 15) / 16;
    int gateBlocks = nblk(T * 32, tb);        // one wave per 16-node tile

    // k = 0: gate on raw features (no norm, no store, no zeroing)
    k_gate<<<gateBlocks, tb, 0, stream>>>((float*)feats, nrm, s,
                                          nullptr, out, N, 0, 0);

    const float* cur = feats;
    for (int k = 0; k < KHOPS; ++k) {
        k_scatter<<<nblk((long)E * 32, tb), tb, 0, stream>>>(cur, nrm, src, dst, E, RAW);
        // gate+accumulate into out, save POST (post-scaled), re-zero RAW
        k_gate<<<gateBlocks, tb, 0, stream>>>(RAW, nrm, s, POST, out, N, 1, 1);
        cur = POST;
    }
}